// WeightIOHMM_81956565943139
// MI455X (gfx1250) — compile-verified
//
#include <hip/hip_runtime.h>

// WeightIOHMM forward-backward on gfx1250 (MI455X), FP32 WMMA 16x16x4.
//
// Problem: 255-step sequential scans (latency bound), working set ~100MB
// (fits 192MB L2). Batch split into 8 chunks of 16 columns; forward and
// backward scans run CONCURRENTLY as 16 blocks of one kernel. Each wave
// holds its 16x256 slice of the transition matrix in 128 VGPRs for the
// whole scan; state double-buffered in LDS (32KB/block).

typedef float v2f __attribute__((ext_vector_type(2)));
typedef float v4f __attribute__((ext_vector_type(4)));
typedef float v8f __attribute__((ext_vector_type(8)));

#define NSTATE 256
#define NBATCH 128
#define SEQLEN 256
#define NLAB   64
#define BC     16   // batch columns per scan block
#define KB     64   // number of K=4 blocks covering NSTATE

static __device__ __forceinline__ v8f wmma_f32(v2f a, v2f b, v8f c) {
  // V_WMMA_F32_16X16X4_F32: D = A(16x4) * B(4x16) + C, fp32 throughout.
  return __builtin_amdgcn_wmma_f32_16x16x4_f32(
      /*neg_a=*/false, a, /*neg_b=*/false, b,
      /*c_mod=*/(short)0, c, /*reuse_a=*/false, /*reuse_b=*/false);
}

// ---------------------------------------------------------------------------
// K1: em[t][b][:] = emb[sentences[b][t]][:]
// ---------------------------------------------------------------------------
__global__ void gather_em(const int* __restrict__ sent,
                          const float* __restrict__ emb,
                          float* __restrict__ em) {
  const int t = blockIdx.x;          // 0..255
  const int b = blockIdx.y;          // 0..127
  const int tok = sent[(size_t)b * SEQLEN + t];
  const v4f* src = (const v4f*)(emb + (size_t)tok * NSTATE);
  v4f* dst = (v4f*)(em + (((size_t)t * NBATCH) + b) * NSTATE);
  dst[threadIdx.x] = src[threadIdx.x];   // 64 threads x float4 = 256 floats
}

// ---------------------------------------------------------------------------
// K2: concurrent scans. blockIdx.x 0..7 -> forward chunk, 8..15 -> backward.
//   fwd[t] = (T   . fwd[t-1]) * em[t]
//   bwd[t] = (T^T . bwd[t+1]) * em[t]
// A-fragments (one 16-row tile of T or T^T per wave) live in VGPRs.
// State (256 states x 16 cols) double-buffered in LDS, layout S[col][state].
// ---------------------------------------------------------------------------
__launch_bounds__(512)
__global__ void scan_pair(const float* __restrict__ Tm,
                          const float* __restrict__ em,
                          float* __restrict__ fwd,
                          float* __restrict__ bwd) {
  __shared__ float S[2][BC * NSTATE];           // 32 KB
  const int tid   = threadIdx.x;
  const int lane  = tid & 31;
  const int w     = tid >> 5;                   // wave = state row-tile 0..15
  const int hf    = lane >> 4;                  // lane half (K/M split)
  const int l16   = lane & 15;
  const int dir   = blockIdx.x >> 3;            // 0 = fwd, 1 = bwd
  const int bbase = (blockIdx.x & 7) * BC;

  // --- A fragments: a[kb] = rows (w*16+l16) of T (fwd) or T^T (bwd),
  //     K = kb*4 + 2*hf + {0,1} per the 16x4 f32 A layout.
  v2f a[KB];
  if (dir == 0) {
    const float* tp = Tm + (size_t)(w * 16 + l16) * NSTATE + 2 * hf;
#pragma unroll
    for (int kb = 0; kb < KB; ++kb) a[kb] = *(const v2f*)(tp + kb * 4);
  } else {
    const float* tp = Tm + (size_t)(2 * hf) * NSTATE + (w * 16 + l16);
#pragma unroll
    for (int kb = 0; kb < KB; ++kb) {
      a[kb].x = tp[(size_t)kb * 4 * NSTATE];
      a[kb].y = tp[(size_t)kb * 4 * NSTATE + NSTATE];
    }
  }

  float* dst = dir ? bwd : fwd;

  // --- init: state(t0) = em[t0]; also emit it to the output array.
  {
    const int t0 = dir ? (SEQLEN - 1) : 0;
    const v4f* e0 = (const v4f*)(em + (((size_t)t0 * NBATCH) + bbase) * NSTATE);
    v4f* g0 = (v4f*)(dst + (((size_t)t0 * NBATCH) + bbase) * NSTATE);
    v4f* s0 = (v4f*)(&S[0][0]);
#pragma unroll
    for (int r = 0; r < 2; ++r) {           // 512 thr * 2 * float4 = 4096 f
      const int i = tid + r * 512;
      const v4f v = e0[i];
      s0[i] = v;
      g0[i] = v;
    }
  }
  __syncthreads();

  for (int step = 1; step < SEQLEN; ++step) {
    const int t   = dir ? (SEQLEN - 1 - step) : step;
    const int cur = (step - 1) & 1, nxt = step & 1;
    // B fragment base: col = l16, state k = kb*4 + 2*hf + v  (contiguous v2f)
    const float* Sb = &S[cur][l16 * NSTATE + 2 * hf];

    v8f acc0 = {0,0,0,0,0,0,0,0}, acc1 = acc0, acc2 = acc0, acc3 = acc0;
#pragma unroll
    for (int kb = 0; kb < KB; kb += 4) {    // 64 WMMAs, 4 independent chains
      const v2f b0 = *(const v2f*)(Sb + (kb + 0) * 4);
      const v2f b1 = *(const v2f*)(Sb + (kb + 1) * 4);
      const v2f b2 = *(const v2f*)(Sb + (kb + 2) * 4);
      const v2f b3 = *(const v2f*)(Sb + (kb + 3) * 4);
      acc0 = wmma_f32(a[kb + 0], b0, acc0);
      acc1 = wmma_f32(a[kb + 1], b1, acc1);
      acc2 = wmma_f32(a[kb + 2], b2, acc2);
      acc3 = wmma_f32(a[kb + 3], b3, acc3);
    }
    const v8f acc = (acc0 + acc1) + (acc2 + acc3);

    // D element (r): state i = w*16 + 8*hf + r, col c = l16 -> 8 consecutive i
    const size_t rowoff =
        (((size_t)t * NBATCH) + bbase + l16) * NSTATE + w * 16 + 8 * hf;
    const v4f e0 = *(const v4f*)(em + rowoff);
    const v4f e1 = *(const v4f*)(em + rowoff + 4);
    const v4f lo = {acc[0] * e0.x, acc[1] * e0.y, acc[2] * e0.z, acc[3] * e0.w};
    const v4f hi = {acc[4] * e1.x, acc[5] * e1.y, acc[6] * e1.z, acc[7] * e1.w};

    float* sp = &S[nxt][l16 * NSTATE + w * 16 + 8 * hf];
    *(v4f*)sp       = lo;
    *(v4f*)(sp + 4) = hi;
    *(v4f*)(dst + rowoff)     = lo;
    *(v4f*)(dst + rowoff + 4) = hi;
    __syncthreads();
  }
}

// ---------------------------------------------------------------------------
// K2b (fallback when ws can't hold 3 arrays): backward scan that fuses
// expected_count in-place:  fx[t] <- (T^T . bwd[t+1]) * fx[t]   (em cancels)
// ---------------------------------------------------------------------------
__launch_bounds__(512)
__global__ void bwd_fused(const float* __restrict__ Tm,
                          const float* __restrict__ em,
                          float* __restrict__ fx) {
  __shared__ float S[2][BC * NSTATE];
  const int tid = threadIdx.x, lane = tid & 31, w = tid >> 5;
  const int hf = lane >> 4, l16 = lane & 15;
  const int bbase = blockIdx.x * BC;

  v2f a[KB];
  const float* tp = Tm + (size_t)(2 * hf) * NSTATE + (w * 16 + l16);
#pragma unroll
  for (int kb = 0; kb < KB; ++kb) {
    a[kb].x = tp[(size_t)kb * 4 * NSTATE];
    a[kb].y = tp[(size_t)kb * 4 * NSTATE + NSTATE];
  }

  {
    const v4f* e0 =
        (const v4f*)(em + (((size_t)(SEQLEN - 1) * NBATCH) + bbase) * NSTATE);
    v4f* s0 = (v4f*)(&S[0][0]);
#pragma unroll
    for (int r = 0; r < 2; ++r) {
      const int i = tid + r * 512;
      s0[i] = e0[i];        // expected[255] == fwd[255]: already in place
    }
  }
  __syncthreads();

  for (int step = 1; step < SEQLEN; ++step) {
    const int t = SEQLEN - 1 - step;
    const int cur = (step - 1) & 1, nxt = step & 1;
    const float* Sb = &S[cur][l16 * NSTATE + 2 * hf];

    v8f acc0 = {0,0,0,0,0,0,0,0}, acc1 = acc0, acc2 = acc0, acc3 = acc0;
#pragma unroll
    for (int kb = 0; kb < KB; kb += 4) {
      const v2f b0 = *(const v2f*)(Sb + (kb + 0) * 4);
      const v2f b1 = *(const v2f*)(Sb + (kb + 1) * 4);
      const v2f b2 = *(const v2f*)(Sb + (kb + 2) * 4);
      const v2f b3 = *(const v2f*)(Sb + (kb + 3) * 4);
      acc0 = wmma_f32(a[kb + 0], b0, acc0);
      acc1 = wmma_f32(a[kb + 1], b1, acc1);
      acc2 = wmma_f32(a[kb + 2], b2, acc2);
      acc3 = wmma_f32(a[kb + 3], b3, acc3);
    }
    const v8f acc = (acc0 + acc1) + (acc2 + acc3);

    const size_t rowoff =
        (((size_t)t * NBATCH) + bbase + l16) * NSTATE + w * 16 + 8 * hf;
    const v4f e0 = *(const v4f*)(em + rowoff);
    const v4f e1 = *(const v4f*)(em + rowoff + 4);
    const v4f f0 = *(const v4f*)(fx + rowoff);
    const v4f f1 = *(const v4f*)(fx + rowoff + 4);
    // backward state carries em factor; expected uses raw matmul result
    const v4f slo = {acc[0] * e0.x, acc[1] * e0.y, acc[2] * e0.z, acc[3] * e0.w};
    const v4f shi = {acc[4] * e1.x, acc[5] * e1.y, acc[6] * e1.z, acc[7] * e1.w};
    const v4f xlo = {acc[0] * f0.x, acc[1] * f0.y, acc[2] * f0.z, acc[3] * f0.w};
    const v4f xhi = {acc[4] * f1.x, acc[5] * f1.y, acc[6] * f1.z, acc[7] * f1.w};

    float* sp = &S[nxt][l16 * NSTATE + w * 16 + 8 * hf];
    *(v4f*)sp       = slo;
    *(v4f*)(sp + 4) = shi;
    *(v4f*)(fx + rowoff)     = xlo;
    *(v4f*)(fx + rowoff + 4) = xhi;
    __syncthreads();
  }
}

// ---------------------------------------------------------------------------
// K3: score[b][t][:] = (fwd[t,b,:]*bwd[t,b,:]/em[t,b,:]) . output
// One (t, batch-chunk) per block; 4 waves, each one 16-wide label tile.
// Output-weight B fragments held in 128 VGPRs per wave.
// ---------------------------------------------------------------------------
__launch_bounds__(128)
__global__ void proj_kernel(const float* __restrict__ fwd,
                            const float* __restrict__ bwd,
                            const float* __restrict__ em,
                            const float* __restrict__ Wout,
                            float* __restrict__ out, int fused) {
  const int t  = blockIdx.x;          // 0..255
  const int bc = blockIdx.y;          // 0..7
  const int tid = threadIdx.x, lane = tid & 31, w = tid >> 5;  // w = n-tile
  const int hf = lane >> 4, l16 = lane & 15;

  // B fragments: Wout[k][n], k = kb*4 + 2*hf + v, n = w*16 + l16
  v2f bw[KB];
  const float* bp = Wout + (size_t)(2 * hf) * NLAB + w * 16 + l16;
#pragma unroll
  for (int kb = 0; kb < KB; ++kb) {
    bw[kb].x = bp[(size_t)kb * 4 * NLAB];
    bw[kb].y = bp[(size_t)kb * 4 * NLAB + NLAB];
  }

  // A fragments: expected[t][bc*16 + m][k], m = l16, k = kb*4 + 2*hf + {0,1}
  const size_t aoff =
      (((size_t)t * NBATCH) + bc * BC + l16) * NSTATE + 2 * hf;

  v8f acc0 = {0,0,0,0,0,0,0,0}, acc1 = acc0, acc2 = acc0, acc3 = acc0;
  if (fused) {   // expected already materialized in `fwd`
#pragma unroll
    for (int kb = 0; kb < KB; kb += 4) {
      const v2f a0 = *(const v2f*)(fwd + aoff + (kb + 0) * 4);
      const v2f a1 = *(const v2f*)(fwd + aoff + (kb + 1) * 4);
      const v2f a2 = *(const v2f*)(fwd + aoff + (kb + 2) * 4);
      const v2f a3 = *(const v2f*)(fwd + aoff + (kb + 3) * 4);
      acc0 = wmma_f32(a0, bw[kb + 0], acc0);
      acc1 = wmma_f32(a1, bw[kb + 1], acc1);
      acc2 = wmma_f32(a2, bw[kb + 2], acc2);
      acc3 = wmma_f32(a3, bw[kb + 3], acc3);
    }
  } else {       // expected = fwd * bwd / em computed on the fly
#pragma unroll
    for (int kb = 0; kb < KB; kb += 2) {
      const v2f f0 = *(const v2f*)(fwd + aoff + (kb + 0) * 4);
      const v2f g0 = *(const v2f*)(bwd + aoff + (kb + 0) * 4);
      const v2f e0 = *(const v2f*)(em  + aoff + (kb + 0) * 4);
      const v2f f1 = *(const v2f*)(fwd + aoff + (kb + 1) * 4);
      const v2f g1 = *(const v2f*)(bwd + aoff + (kb + 1) * 4);
      const v2f e1 = *(const v2f*)(em  + aoff + (kb + 1) * 4);
      const v2f a0 = {f0.x * g0.x / e0.x, f0.y * g0.y / e0.y};
      const v2f a1 = {f1.x * g1.x / e1.x, f1.y * g1.y / e1.y};
      acc0 = wmma_f32(a0, bw[kb + 0], acc0);
      acc1 = wmma_f32(a1, bw[kb + 1], acc1);
    }
  }
  const v8f acc = (acc0 + acc1) + (acc2 + acc3);

  // D: batch m = 8*hf + r, label n = w*16 + l16 ; out layout (B, T, N)
#pragma unroll
  for (int r = 0; r < 8; ++r) {
    const size_t b = (size_t)(bc * BC + 8 * hf + r);
    out[(b * SEQLEN + t) * NLAB + w * 16 + l16] = acc[r];
  }
}

// ---------------------------------------------------------------------------
extern "C" void kernel_launch(void* const* d_in, const int* in_sizes, int n_in,
                              void* d_out, int out_size, void* d_ws,
                              size_t ws_size, hipStream_t stream) {
  const int*   sent = (const int*)d_in[0];    // (128, 256)
  const float* emb  = (const float*)d_in[1];  // (32000, 256)
  const float* Tm   = (const float*)d_in[2];  // (256, 256)
  const float* Wout = (const float*)d_in[3];  // (256, 64)
  float* out = (float*)d_out;                 // (128, 256, 64)

  const size_t NE = (size_t)SEQLEN * NBATCH * NSTATE;  // 8.4M floats
  float* em  = (float*)d_ws;
  float* fwd = em + NE;

  gather_em<<<dim3(SEQLEN, NBATCH), 64, 0, stream>>>(sent, emb, em);

  if (ws_size >= 3 * NE * sizeof(float)) {
    // Concurrent forward+backward scans (halves the sequential critical path)
    float* bwd = fwd + NE;
    scan_pair<<<16, 512, 0, stream>>>(Tm, em, fwd, bwd);
    proj_kernel<<<dim3(SEQLEN, 8), 128, 0, stream>>>(fwd, bwd, em, Wout, out, 0);
  } else {
    // Serial fallback: forward scan, then backward scan fusing expected_count
    scan_pair<<<8, 512, 0, stream>>>(Tm, em, fwd, nullptr);
    bwd_fused<<<8, 512, 0, stream>>>(Tm, em, fwd);
    proj_kernel<<<dim3(SEQLEN, 8), 128, 0, stream>>>(fwd, fwd, em, Wout, out, 1);
  }
}